// GCN_36378372997641
// MI455X (gfx1250) — compile-verified
//
#include <hip/hip_runtime.h>
#include <hip/hip_bf16.h>
#include <math.h>

typedef float v2f __attribute__((ext_vector_type(2)));
typedef float v8f __attribute__((ext_vector_type(8)));

#define N_NODES_C 50000
#define N_EDGES_C 800000
#define N_GRAPHS_C 512
#define DIM_C 256
#define OUT_DIM_C 10

// ---------------------------------------------------------------------------
// Degree: deg[dst] += 1  (f32 atomics; deg buffer pre-zeroed)
// ---------------------------------------------------------------------------
__global__ void gcn_deg_kernel(const int* __restrict__ dst, float* __restrict__ deg, int nE) {
    int e = blockIdx.x * blockDim.x + threadIdx.x;
    if (e < nE) atomicAdd(&deg[dst[e]], 1.0f);
}

// dis[i] = rsqrt(1 + deg[i])   (in place)
__global__ void gcn_dis_kernel(float* __restrict__ deg, int n) {
    int i = blockIdx.x * blockDim.x + threadIdx.x;
    if (i < n) deg[i] = rsqrtf(1.0f + deg[i]);
}

// ---------------------------------------------------------------------------
// WMMA GEMM: C[M x 256] = A[M x 256] * W[256 x 256]
// Block = 128 threads = 4 wave32. Block owns one 16-row M tile (M % 16 == 0).
// A panel (16x256 f32, 16KB) staged in LDS; each wave computes 4 N-tiles of 16
// via V_WMMA_F32_16X16X4_F32, K-loop stepping 4.
//
// A 16x4 layout: lanes 0-15 M=0..15; VGPR0 = K=(0|2) by half, VGPR1 = K=(1|3).
// B 4x16 layout: rows striped across lanes per VGPR, mirrored halves.
// C/D 16x16:     VGPR j -> row j (lanes 0-15) / row 8+j (lanes 16-31), col = lane%16.
// ---------------------------------------------------------------------------
__global__ __launch_bounds__(128) void gcn_gemm_wmma_f32(
    const float* __restrict__ A, const float* __restrict__ W,
    float* __restrict__ C, int M) {
    __shared__ float lA[16 * DIM_C];           // 16KB of the 320KB/WGP LDS

    const int tid   = threadIdx.x;
    const int mtile = blockIdx.x;              // 16 rows per tile, exact (50000/16)
    const float* Arow = A + (size_t)mtile * 16 * DIM_C;

    #pragma unroll 4
    for (int i = tid; i < 16 * DIM_C; i += 128) lA[i] = Arow[i];
    __syncthreads();

    const int wave = tid >> 5;
    const int lane = tid & 31;
    const int half = lane >> 4;                // 0: lanes 0-15, 1: lanes 16-31
    const int lid  = lane & 15;

    for (int nt = 0; nt < 4; ++nt) {
        const int col0 = (wave * 4 + nt) * 16;
        v8f c = {};
        for (int k = 0; k < DIM_C; k += 4) {
            const int kk = k + 2 * half;
            v2f a, b;
            a.x = lA[lid * DIM_C + kk];
            a.y = lA[lid * DIM_C + kk + 1];
            b.x = W[(size_t)kk * DIM_C + col0 + lid];
            b.y = W[(size_t)(kk + 1) * DIM_C + col0 + lid];
            // D = A x B + C  (exact f32 path; 8-arg form: neg_a,A,neg_b,B,c_mod,C,reuse_a,reuse_b)
            c = __builtin_amdgcn_wmma_f32_16x16x4_f32(
                    false, a, false, b, (short)0, c, false, false);
        }
        float* Cout = C + ((size_t)mtile * 16) * DIM_C + col0;
        #pragma unroll
        for (int j = 0; j < 8; ++j) {
            Cout[(size_t)(j + 8 * half) * DIM_C + lid] = c[j];
        }
    }
}

// ---------------------------------------------------------------------------
// Edge scatter: agg[dst] += h[src] * (dis[src]*dis[dst])
// 64 lanes per edge, float4 per lane -> coalesced 1KB gather, f32 atomics into
// a 51.2MB buffer that is resident in the 192MB L2.
// ---------------------------------------------------------------------------
__global__ void gcn_edge_scatter(const float* __restrict__ h,
                                 const int* __restrict__ src,
                                 const int* __restrict__ dst,
                                 const float* __restrict__ dis,
                                 float* __restrict__ agg, int nE) {
    int t  = blockIdx.x * blockDim.x + threadIdx.x;
    int e  = t >> 6;
    int f4 = t & 63;
    if (e >= nE) return;
    const int s = src[e];
    const int d = dst[e];
    const float norm = dis[s] * dis[d];
    float4 v = ((const float4*)(h + (size_t)s * DIM_C))[f4];
    float* out = agg + (size_t)d * DIM_C + f4 * 4;
    atomicAdd(out + 0, v.x * norm);
    atomicAdd(out + 1, v.y * norm);
    atomicAdd(out + 2, v.z * norm);
    atomicAdd(out + 3, v.w * norm);
}

// agg[i] = relu(agg[i] + h[i]*dis[i]^2 + b)   (fused self-loop + bias + relu)
__global__ void gcn_self_relu(float* __restrict__ agg, const float* __restrict__ h,
                              const float* __restrict__ dis, const float* __restrict__ b,
                              int n) {
    int t  = blockIdx.x * blockDim.x + threadIdx.x;
    int i  = t >> 6;
    int f4 = t & 63;
    if (i >= n) return;
    const float d2 = dis[i] * dis[i];
    float4 a  = ((const float4*)(agg + (size_t)i * DIM_C))[f4];
    float4 hv = ((const float4*)(h + (size_t)i * DIM_C))[f4];
    float4 bv = ((const float4*)b)[f4];
    a.x = fmaxf(fmaf(hv.x, d2, a.x) + bv.x, 0.0f);
    a.y = fmaxf(fmaf(hv.y, d2, a.y) + bv.y, 0.0f);
    a.z = fmaxf(fmaf(hv.z, d2, a.z) + bv.z, 0.0f);
    a.w = fmaxf(fmaf(hv.w, d2, a.w) + bv.w, 0.0f);
    ((float4*)(agg + (size_t)i * DIM_C))[f4] = a;
}

// Mean-pool accumulation: sums[batch[i]] += h[i]; cnts[batch[i]] += 1
__global__ void gcn_pool(const float* __restrict__ h, const int* __restrict__ batch,
                         float* __restrict__ sums, float* __restrict__ cnts, int n) {
    int t  = blockIdx.x * blockDim.x + threadIdx.x;
    int i  = t >> 6;
    int f4 = t & 63;
    if (i >= n) return;
    const int g = batch[i];
    float4 v = ((const float4*)(h + (size_t)i * DIM_C))[f4];
    float* out = sums + (size_t)g * DIM_C + f4 * 4;
    atomicAdd(out + 0, v.x);
    atomicAdd(out + 1, v.y);
    atomicAdd(out + 2, v.z);
    atomicAdd(out + 3, v.w);
    if (f4 == 0) atomicAdd(&cnts[g], 1.0f);
}

// Per-graph head: pooled = sums/cnt; logits = pooled @ Wf + bf; log_softmax.
__global__ __launch_bounds__(256) void gcn_head(const float* __restrict__ sums,
                                                const float* __restrict__ cnts,
                                                const float* __restrict__ Wf,
                                                const float* __restrict__ bf,
                                                float* __restrict__ out) {
    __shared__ float red[256];
    __shared__ float logits[16];
    const int g = blockIdx.x;
    const int k = threadIdx.x;
    const float cnt = fmaxf(cnts[g], 1.0f);
    const float p = sums[(size_t)g * DIM_C + k] / cnt;
    for (int o = 0; o < OUT_DIM_C; ++o) {
        red[k] = p * Wf[(size_t)k * OUT_DIM_C + o];
        __syncthreads();
        for (int s = 128; s > 0; s >>= 1) {
            if (k < s) red[k] += red[k + s];
            __syncthreads();
        }
        if (k == 0) logits[o] = red[0] + bf[o];
        __syncthreads();
    }
    if (k == 0) {
        float m = -INFINITY;
        for (int o = 0; o < OUT_DIM_C; ++o) m = fmaxf(m, logits[o]);
        float sse = 0.0f;
        for (int o = 0; o < OUT_DIM_C; ++o) sse += expf(logits[o] - m);
        const float lse = logf(sse);
        for (int o = 0; o < OUT_DIM_C; ++o) out[(size_t)g * OUT_DIM_C + o] = logits[o] - m - lse;
    }
}

static inline size_t align256(size_t x) { return (x + 255) & ~size_t(255); }

extern "C" void kernel_launch(void* const* d_in, const int* in_sizes, int n_in,
                              void* d_out, int out_size, void* d_ws, size_t ws_size,
                              hipStream_t stream) {
    const float* x     = (const float*)d_in[0];
    const int*   edge  = (const int*)d_in[1];
    const int*   batch = (const int*)d_in[2];
    const float* W1    = (const float*)d_in[3];
    const float* b1    = (const float*)d_in[4];
    const float* W2    = (const float*)d_in[5];
    const float* b2    = (const float*)d_in[6];
    const float* Wf    = (const float*)d_in[7];
    const float* bf    = (const float*)d_in[8];

    const int n  = in_sizes[2];          // 50000 nodes
    const int nE = in_sizes[1] / 2;      // 800000 edges
    const int* src = edge;
    const int* dst = edge + nE;

    // Workspace layout
    char* ws = (char*)d_ws;
    size_t off = 0;
    float* dis  = (float*)(ws + off); off += align256((size_t)n * sizeof(float));
    float* bufH = (float*)(ws + off); off += align256((size_t)n * DIM_C * sizeof(float));
    float* bufA = (float*)(ws + off); off += align256((size_t)n * DIM_C * sizeof(float));
    float* sums = (float*)(ws + off); off += align256((size_t)N_GRAPHS_C * DIM_C * sizeof(float));
    float* cnts = (float*)(ws + off); off += align256((size_t)N_GRAPHS_C * sizeof(float));
    (void)ws_size;

    const size_t featBytes = (size_t)n * DIM_C * sizeof(float);
    const int mtiles      = n / 16;                       // 3125, exact
    const int edgeBlocks  = (nE * 64 + 255) / 256;        // 64 lanes/edge
    const int nodeBlocks  = (n * 64 + 255) / 256;

    // ---- normalization ----
    hipMemsetAsync(dis, 0, (size_t)n * sizeof(float), stream);
    gcn_deg_kernel<<<(nE + 255) / 256, 256, 0, stream>>>(dst, dis, nE);
    gcn_dis_kernel<<<(n + 255) / 256, 256, 0, stream>>>(dis, n);

    // ---- layer 1: h = x@W1 ; agg = scatter + self ; relu ----
    hipMemsetAsync(bufA, 0, featBytes, stream);
    gcn_gemm_wmma_f32<<<mtiles, 128, 0, stream>>>(x, W1, bufH, n);
    gcn_edge_scatter<<<edgeBlocks, 256, 0, stream>>>(bufH, src, dst, dis, bufA, nE);
    gcn_self_relu<<<nodeBlocks, 256, 0, stream>>>(bufA, bufH, dis, b1, n);

    // ---- layer 2: h = relu1@W2 ; agg ; relu ----
    gcn_gemm_wmma_f32<<<mtiles, 128, 0, stream>>>(bufA, W2, bufH, n);
    hipMemsetAsync(bufA, 0, featBytes, stream);
    gcn_edge_scatter<<<edgeBlocks, 256, 0, stream>>>(bufH, src, dst, dis, bufA, nE);
    gcn_self_relu<<<nodeBlocks, 256, 0, stream>>>(bufA, bufH, dis, b2, n);

    // ---- mean pool + head ----
    hipMemsetAsync(sums, 0, (size_t)N_GRAPHS_C * DIM_C * sizeof(float), stream);
    hipMemsetAsync(cnts, 0, (size_t)N_GRAPHS_C * sizeof(float), stream);
    gcn_pool<<<nodeBlocks, 256, 0, stream>>>(bufA, batch, sums, cnts, n);
    gcn_head<<<N_GRAPHS_C, 256, 0, stream>>>(sums, cnts, Wf, bf, (float*)d_out);
}